// GAT_80092550135830
// MI455X (gfx1250) — compile-verified
//
#include <hip/hip_runtime.h>
#include <math.h>

// ---------------------------------------------------------------------------
// HAN (heterogeneous graph attention) forward for MI455X / gfx1250.
// fp32 everywhere (matches reference numerics); GEMMs via V_WMMA_F32_16X16X4_F32.
// B (weight) fragments are register-cached and reused across RT row tiles.
// ---------------------------------------------------------------------------

typedef float v2f __attribute__((ext_vector_type(2)));
typedef float v8f __attribute__((ext_vector_type(8)));

#define DH 128
#define NHEAD 8
#define HD 16
#define RT 5  // row tiles per wave; 3125 and 1250 are both divisible by 5

// ---------------------------------------------------------------------------
// GEMM: Y[N,128] = X[N,128] @ W[128,128] + bias[128]
// One wave computes RT 16x16 tiles of one column block. grid = (rowTiles/RT, 8).
// A-frag (16x4 f32): lanes 0-15 -> K={0,1}, lanes 16-31 -> K={2,3}, M = lane&15.
// B-frag (4x16 f32): same K split, N = lane&15.  (cached in 64 VGPRs, K=0..127)
// C/D (16x16 f32, 8 VGPRs): vgpr r -> M = r + 8*(lane>=16), N = lane&15.
// ---------------------------------------------------------------------------
__global__ __launch_bounds__(32) void gemm128_wmma(
    const float* __restrict__ X, const float* __restrict__ W,
    const float* __restrict__ bias, float* __restrict__ Y, int rowTiles) {
  const int rt0 = blockIdx.x * RT;
  const int ct = blockIdx.y;
  const int lane = threadIdx.x;
  const int half = lane >> 4;
  const int l16 = lane & 15;
  const int col = ct * 16 + l16;
  const int kb = half * 2;

  v2f bfrag[32];
#pragma unroll
  for (int i = 0; i < 32; ++i) {
    const int k = i * 4 + kb;
    bfrag[i].x = W[(size_t)k * DH + col];
    bfrag[i].y = W[(size_t)(k + 1) * DH + col];
  }
  const float bv = bias[col];

  for (int t = 0; t < RT; ++t) {
    const int rt = rt0 + t;
    if (rt >= rowTiles) break;  // wave-uniform guard: EXEC stays all-ones
    const float* xrow = X + (size_t)(rt * 16 + l16) * DH;
    v8f acc = {};
#pragma unroll
    for (int i = 0; i < 32; ++i) {
      v2f a;
      a.x = xrow[i * 4 + kb];
      a.y = xrow[i * 4 + kb + 1];
      acc = __builtin_amdgcn_wmma_f32_16x16x4_f32(false, a, false, bfrag[i],
                                                  (short)0, acc, false, false);
    }
    const int mbase = rt * 16 + half * 8;
#pragma unroll
    for (int r = 0; r < 8; ++r)
      Y[(size_t)(mbase + r) * DH + col] = acc[r] + bv;
  }
}

// ---------------------------------------------------------------------------
// Semantic-attention score: accum += sum_{n,d} q[d] * tanh((relu(A)@Kw)[n,d] + kb[d])
// Same tiling/B-caching; relu fused into A-fragment load; per-lane tanh reduce
// across RT tiles, one wave reduction + one atomicAdd per wave.
// ---------------------------------------------------------------------------
__global__ __launch_bounds__(32) void score128_wmma(
    const float* __restrict__ A, const float* __restrict__ Kw,
    const float* __restrict__ kb, const float* __restrict__ q,
    float* __restrict__ accum, int rowTiles) {
  const int rt0 = blockIdx.x * RT;
  const int ct = blockIdx.y;
  const int lane = threadIdx.x;
  const int half = lane >> 4;
  const int l16 = lane & 15;
  const int col = ct * 16 + l16;
  const int kbase = half * 2;

  v2f bfrag[32];
#pragma unroll
  for (int i = 0; i < 32; ++i) {
    const int k = i * 4 + kbase;
    bfrag[i].x = Kw[(size_t)k * DH + col];
    bfrag[i].y = Kw[(size_t)(k + 1) * DH + col];
  }
  const float qc = q[col], kc = kb[col];

  float part = 0.f;
  for (int t = 0; t < RT; ++t) {
    const int rt = rt0 + t;
    if (rt >= rowTiles) break;  // wave-uniform
    const float* xrow = A + (size_t)(rt * 16 + l16) * DH;
    v8f acc = {};
#pragma unroll
    for (int i = 0; i < 32; ++i) {
      v2f a;
      a.x = fmaxf(xrow[i * 4 + kbase], 0.f);
      a.y = fmaxf(xrow[i * 4 + kbase + 1], 0.f);
      acc = __builtin_amdgcn_wmma_f32_16x16x4_f32(false, a, false, bfrag[i],
                                                  (short)0, acc, false, false);
    }
#pragma unroll
    for (int r = 0; r < 8; ++r) part += qc * tanhf(acc[r] + kc);
  }
#pragma unroll
  for (int off = 16; off > 0; off >>= 1) part += __shfl_xor(part, off, 32);
  if (lane == 0) atomicAdd(accum, part);
}

// ---------------------------------------------------------------------------
// Per-node attention coefficients: out[n,h] = sum_d X[n, h*16+d] * att[h*16+d]
// ---------------------------------------------------------------------------
__global__ __launch_bounds__(256) void att_dot(
    const float* __restrict__ X, const float* __restrict__ att,
    float* __restrict__ out, int n8) {
  int t = blockIdx.x * 256 + threadIdx.x;
  if (t >= n8) return;
  const int n = t >> 3, h = t & 7;
  const float4* x4 = (const float4*)(X + (size_t)n * DH + h * HD);
  const float4* a4 = (const float4*)(att + h * HD);
  float s = 0.f;
#pragma unroll
  for (int i = 0; i < 4; ++i) {
    float4 xv = x4[i], av = a4[i];
    s += xv.x * av.x + xv.y * av.y + xv.z * av.z + xv.w * av.w;
  }
  out[t] = s;
}

__device__ __forceinline__ float lrelu(float x) { return x > 0.f ? x : 0.2f * x; }

// segment max via atomic bit tricks (m pre-initialized to 0xFFFFFFFF per byte)
__global__ __launch_bounds__(256) void edge_max_k(
    const int* __restrict__ src, const int* __restrict__ dst,
    const float* __restrict__ as, const float* __restrict__ ad,
    float* __restrict__ m, int e8) {
  int t = blockIdx.x * 256 + threadIdx.x;
  if (t >= e8) return;
  const int e = t >> 3, h = t & 7;
  const int si = src[e], di = dst[e];
  const float v = lrelu(as[si * NHEAD + h] + ad[di * NHEAD + h]);
  float* addr = m + (size_t)di * NHEAD + h;
  if (v >= 0.f)
    atomicMax((int*)addr, __float_as_int(v));
  else
    atomicMin((unsigned int*)addr, __float_as_uint(v));
}

__global__ __launch_bounds__(256) void edge_sum_k(
    const int* __restrict__ src, const int* __restrict__ dst,
    const float* __restrict__ as, const float* __restrict__ ad,
    const float* __restrict__ m, float* __restrict__ ssum, int e8) {
  int t = blockIdx.x * 256 + threadIdx.x;
  if (t >= e8) return;
  const int e = t >> 3, h = t & 7;
  const int si = src[e], di = dst[e];
  const size_t idx = (size_t)di * NHEAD + h;
  const float v = lrelu(as[si * NHEAD + h] + ad[di * NHEAD + h]);
  atomicAdd(&ssum[idx], __expf(v - m[idx]));
}

__global__ __launch_bounds__(256) void edge_agg_k(
    const int* __restrict__ src, const int* __restrict__ dst,
    const float* __restrict__ as, const float* __restrict__ ad,
    const float* __restrict__ m, const float* __restrict__ ssum,
    const float* __restrict__ H, float* __restrict__ agg, int e8) {
  int t = blockIdx.x * 256 + threadIdx.x;
  if (t >= e8) return;
  const int e = t >> 3, h = t & 7;
  const int si = src[e], di = dst[e];
  const size_t idx = (size_t)di * NHEAD + h;
  const float v = lrelu(as[si * NHEAD + h] + ad[di * NHEAD + h]);
  const float w = __expf(v - m[idx]) / (ssum[idx] + 1e-16f);
  const float* hs = H + (size_t)si * DH + h * HD;
  float* ag = agg + (size_t)di * DH + h * HD;
#pragma unroll
  for (int d = 0; d < HD; ++d) atomicAdd(&ag[d], hs[d] * w);
}

// semantic softmax over 2 relations (protein); logits are mean-scaled scores
__global__ void beta_k(const float* __restrict__ scores, float* __restrict__ beta,
                       float invN) {
  const float s0 = scores[0] * invN, s1 = scores[1] * invN;
  const float mm = fmaxf(s0, s1);
  const float e0 = __expf(s0 - mm), e1 = __expf(s1 - mm);
  const float inv = 1.f / (e0 + e1);
  beta[0] = e0 * inv;
  beta[1] = e1 * inv;
}

__global__ __launch_bounds__(256) void combine2_k(
    const float* __restrict__ a0, const float* __restrict__ a1,
    const float* __restrict__ beta, float* __restrict__ out, int n) {
  int t = blockIdx.x * 256 + threadIdx.x;
  if (t >= n) return;
  out[t] = beta[0] * fmaxf(a0[t], 0.f) + beta[1] * fmaxf(a1[t], 0.f);
}

__global__ __launch_bounds__(256) void combine1_k(
    const float* __restrict__ a0, float* __restrict__ out, int n) {
  int t = blockIdx.x * 256 + threadIdx.x;
  if (t >= n) return;
  out[t] = fmaxf(a0[t], 0.f);  // single relation => beta == 1
}

// ---------------------------------------------------------------------------
// Host orchestration
// Input order (jax tree_flatten, dict keys sorted):
//  0: x_protein  1: x_go
//  2..40: 3 layers x 13 leaves:
//    +0 att.ann.dst +1 att.ann.src +2 att.ppi.dst +3 att.ppi.src
//    +4 att.rev.dst +5 att.rev.src +6 k_b +7 k_w
//    +8 proj.go.b +9 proj.go.w +10 proj.prot.b +11 proj.prot.w +12 q
//  41: lin.b  42: lin.w
//  43..48: ppi_src, ppi_dst, ann_src, ann_dst, rev_src, rev_dst
// ---------------------------------------------------------------------------
extern "C" void kernel_launch(void* const* d_in, const int* in_sizes, int n_in,
                              void* d_out, int out_size, void* d_ws, size_t ws_size,
                              hipStream_t stream) {
  const int NP = in_sizes[0] / DH;   // 50000
  const int NG = in_sizes[1] / DH;   // 20000
  const int E  = in_sizes[43];       // 600000
  const int E8 = E * NHEAD;
  const int rtP = NP / 16, rtG = NG / 16;  // 3125, 1250 (exact)

  auto P = [&](int layer, int leaf) -> const float* {
    return (const float*)d_in[2 + layer * 13 + leaf];
  };
  const float* lin_b = (const float*)d_in[41];
  const float* lin_w = (const float*)d_in[42];
  const int* ppi_s = (const int*)d_in[43];
  const int* ppi_d = (const int*)d_in[44];
  const int* ann_s = (const int*)d_in[45];
  const int* ann_d = (const int*)d_in[46];
  const int* rev_s = (const int*)d_in[47];
  const int* rev_d = (const int*)d_in[48];

  const size_t np128 = (size_t)NP * DH, ng128 = (size_t)NG * DH;
  float* ws = (float*)d_ws;
  float* x_prot  = ws;                         ws += np128;
  float* x_go    = ws;                         ws += ng128;
  float* h_prot  = ws;                         ws += np128;
  float* h_go    = ws;                         ws += ng128;
  float* agg_ppi = ws;                         ws += np128;   // protein relation 0
  float* agg_rev = ws;                         ws += np128;   // protein relation 1
  float* agg_ann = ws;                         ws += ng128;   // go relation 0
  float* a_src   = ws;                         ws += (size_t)NP * NHEAD;
  float* a_dst   = ws;                         ws += (size_t)NP * NHEAD;
  float* segmax  = ws;                         ws += (size_t)NP * NHEAD;
  float* segsum  = ws;                         ws += (size_t)NP * NHEAD;
  float* scores  = ws;                         ws += 2;
  float* beta    = ws;                         ws += 2;

  auto cdiv = [](long long a, int b) { return (int)((a + b - 1) / b); };
  const dim3 gP(cdiv(rtP, RT), DH / 16), gG(cdiv(rtG, RT), DH / 16);
  const int bT = 256;

  auto run_edge = [&](const int* es, const int* ed, const float* hs,
                      const float* hd, const float* att_s, const float* att_d,
                      float* agg, int Ns, int Nd) {
    att_dot<<<cdiv((long long)Ns * NHEAD, bT), bT, 0, stream>>>(hs, att_s, a_src, Ns * NHEAD);
    att_dot<<<cdiv((long long)Nd * NHEAD, bT), bT, 0, stream>>>(hd, att_d, a_dst, Nd * NHEAD);
    hipMemsetAsync(segmax, 0xFF, (size_t)Nd * NHEAD * sizeof(float), stream);
    hipMemsetAsync(segsum, 0, (size_t)Nd * NHEAD * sizeof(float), stream);
    edge_max_k<<<cdiv(E8, bT), bT, 0, stream>>>(es, ed, a_src, a_dst, segmax, E8);
    edge_sum_k<<<cdiv(E8, bT), bT, 0, stream>>>(es, ed, a_src, a_dst, segmax, segsum, E8);
    edge_agg_k<<<cdiv(E8, bT), bT, 0, stream>>>(es, ed, a_src, a_dst, segmax, segsum, hs, agg, E8);
  };

  for (int l = 0; l < 3; ++l) {
    const float* xp = (l == 0) ? (const float*)d_in[0] : x_prot;
    const float* xg = (l == 0) ? (const float*)d_in[1] : x_go;

    // node-type projections (WMMA GEMM + bias)
    gemm128_wmma<<<gP, 32, 0, stream>>>(xp, P(l, 11), P(l, 10), h_prot, rtP);
    gemm128_wmma<<<gG, 32, 0, stream>>>(xg, P(l, 9),  P(l, 8),  h_go,   rtG);

    // zero the three relation accumulators (contiguous block)
    hipMemsetAsync(agg_ppi, 0, (2 * np128 + ng128) * sizeof(float), stream);

    // ppi: protein -> protein ; ann: protein -> go ; rev: go -> protein
    run_edge(ppi_s, ppi_d, h_prot, h_prot, P(l, 3), P(l, 2), agg_ppi, NP, NP);
    run_edge(ann_s, ann_d, h_prot, h_go,   P(l, 1), P(l, 0), agg_ann, NP, NG);
    run_edge(rev_s, rev_d, h_go,   h_prot, P(l, 5), P(l, 4), agg_rev, NG, NP);

    // semantic attention (protein has 2 relations; go has 1 => beta = 1)
    hipMemsetAsync(scores, 0, 2 * sizeof(float), stream);
    score128_wmma<<<gP, 32, 0, stream>>>(agg_ppi, P(l, 7), P(l, 6), P(l, 12), &scores[0], rtP);
    score128_wmma<<<gP, 32, 0, stream>>>(agg_rev, P(l, 7), P(l, 6), P(l, 12), &scores[1], rtP);
    beta_k<<<1, 1, 0, stream>>>(scores, beta, 1.0f / (float)NP);
    combine2_k<<<cdiv((long long)np128, bT), bT, 0, stream>>>(agg_ppi, agg_rev, beta, x_prot, (int)np128);
    combine1_k<<<cdiv((long long)ng128, bT), bT, 0, stream>>>(agg_ann, x_go, (int)ng128);
    // inter-layer ReLU is a no-op: convex combination of ReLU'd tensors >= 0
  }

  // final shared linear layer into d_out (protein block, then go block)
  float* out = (float*)d_out;
  gemm128_wmma<<<gP, 32, 0, stream>>>(x_prot, lin_w, lin_b, out, rtP);
  gemm128_wmma<<<gG, 32, 0, stream>>>(x_go, lin_w, lin_b, out + np128, rtG);
}